// Layer_1382979469616
// MI455X (gfx1250) — compile-verified
//
#include <hip/hip_runtime.h>

#define IN_DIM      512
#define OUT_DIM     256
#define MAX_OUT_SEQ 512
#define SEQ         1024
#define BATCH       2
#define OUT_SEQ     512          // SEQ / 2
#define TOTAL_ROWS  2048         // BATCH * SEQ
#define LN_EPS      1e-5f

typedef __attribute__((ext_vector_type(2))) float v2f;
typedef __attribute__((ext_vector_type(8))) float v8f;

// ---------------------------------------------------------------------------
// Kernel A: Z = x @ M^T   (2048x512 @ 512x256), f32 WMMA 16x16x4.
// One wave per 16x16 C tile; A/B fragment lane layout per CDNA5 ISA 7.12.2:
//   lanes 0-15 hold K=0,1 in v0,v1 ; lanes 16-31 hold K=2,3.
// Both A (x row-major) and B (= M^T, i.e. rows of M) give each lane two
// consecutive floats -> single 8-byte load per fragment.
// ---------------------------------------------------------------------------
__global__ __launch_bounds__(256) void gemm_xMt(const float* __restrict__ x,
                                                const float* __restrict__ M,
                                                float* __restrict__ Z) {
  const int lane = threadIdx.x & 31;
  const int wave = threadIdx.x >> 5;
  const int tile = blockIdx.x * 8 + wave;     // 2048 tiles total
  const int tN = tile & 15;                   // 256/16 = 16 tiles along N
  const int tM = tile >> 4;                   // 2048/16 = 128 tiles along M
  const int m0 = tM * 16, n0 = tN * 16;
  const int half = lane >> 4;                 // K sub-pair selector
  const int l = lane & 15;

  v8f c = {};
  const float* arow = x + (size_t)(m0 + l) * IN_DIM + 2 * half;
  const float* brow = M + (size_t)(n0 + l) * IN_DIM + 2 * half;
#pragma unroll 8
  for (int kc = 0; kc < IN_DIM; kc += 4) {
    v2f a = *(const v2f*)(arow + kc);
    v2f b = *(const v2f*)(brow + kc);
    c = __builtin_amdgcn_wmma_f32_16x16x4_f32(false, a, false, b,
                                              (short)0, c, false, false);
  }
  // C/D layout: VGPR vg -> row m0+vg (lanes 0-15) / m0+8+vg (lanes 16-31)
  float* out = Z + (size_t)(m0 + half * 8) * OUT_DIM + n0 + l;
#pragma unroll
  for (int vg = 0; vg < 8; ++vg)
    out[(size_t)vg * OUT_DIM] = c[vg];
}

// ---------------------------------------------------------------------------
// Kernel B: per-row LayerNorm + phi-modulated contraction + residual fold.
//   S[k,i] = Z[k,i] + sum_j Zn[k,j] * P[i,j] * cos(2*pi*k / (i*256+j+2))
// One 256-thread block (8 wave32) per row k; thread t owns output column i=t.
// Transcendental-bound: v_cos_f32 computes cos(2*pi*x), so the argument is
// simply k * rcp(period) -- no 2*pi scaling needed.
// ---------------------------------------------------------------------------
__global__ __launch_bounds__(256) void modrow(const float* __restrict__ Z,
                                              const float* __restrict__ P,
                                              const float* __restrict__ gamma,
                                              const float* __restrict__ beta,
                                              float* __restrict__ S) {
  __shared__ float zn[OUT_DIM];
  __shared__ float red[16];
  const int k = blockIdx.x;
  const int t = threadIdx.x;
  const float z = Z[(size_t)k * OUT_DIM + t];

  // mean / var reduction: wave32 shuffle tree, then cross-wave via LDS
  float s = z, sq = z * z;
#pragma unroll
  for (int off = 16; off > 0; off >>= 1) {
    s  += __shfl_xor(s, off, 32);
    sq += __shfl_xor(sq, off, 32);
  }
  if ((t & 31) == 0) { red[t >> 5] = s; red[8 + (t >> 5)] = sq; }
  __syncthreads();
  if (t == 0) {
    float ts = 0.f, tq = 0.f;
#pragma unroll
    for (int w = 0; w < 8; ++w) { ts += red[w]; tq += red[8 + w]; }
    const float mu = ts * (1.0f / OUT_DIM);
    red[0] = mu;
    red[1] = __builtin_amdgcn_rsqf(tq * (1.0f / OUT_DIM) - mu * mu + LN_EPS);
  }
  __syncthreads();
  const float mu = red[0], inv = red[1];
  zn[t] = (z - mu) * inv * gamma[t] + beta[t];
  __syncthreads();

  const float kf    = (float)k;
  const float pbase = (float)(t * OUT_DIM + 2);   // exact in f32 (<= 65793)
  const float* prow = P + (size_t)t * OUT_DIM;
  float acc = 0.f;
#pragma unroll 4
  for (int j = 0; j < OUT_DIM; ++j) {
    const float p = pbase + (float)j;
    // V_COS_F32(x) = cos(2*pi*x)  ->  cos(2*pi*k/p)
    const float c = __builtin_amdgcn_cosf(kf * __builtin_amdgcn_rcpf(p));
    acc = __builtin_fmaf(zn[j] * prow[j], c, acc);
  }
  S[(size_t)k * OUT_DIM + t] = acc + z;   // U+R folded: S = T + Z
}

// ---------------------------------------------------------------------------
// Kernel C: out[b] = AL^T @ S_b  (512x1024 @ 1024x256) per batch, f32 WMMA.
// A[t,s] = Linker[s*MAX_OUT_SEQ + t]; fragment elements are column-strided so
// each lane issues two b32 loads per side (all L2-resident; GEMM is tiny).
// ---------------------------------------------------------------------------
__global__ __launch_bounds__(256) void gemm_link(const float* __restrict__ Lnk,
                                                 const float* __restrict__ S,
                                                 float* __restrict__ out,
                                                 int out_size) {
  const int lane = threadIdx.x & 31;
  const int wave = threadIdx.x >> 5;
  const int tile = blockIdx.x * 8 + wave;   // 1024 tiles total
  const int b  = tile >> 9;                 // 512 tiles per batch
  const int r  = tile & 511;
  const int tN = r & 15;                    // o tiles: 256/16
  const int tT = r >> 4;                    // t tiles: 512/16 = 32
  const int t0 = tT * 16, o0 = tN * 16;
  const int half = lane >> 4;
  const int l = lane & 15;

  v8f c = {};
  const float* abase = Lnk + t0 + l;                              // col t0+l
  const float* bbase = S + (size_t)(b * SEQ) * OUT_DIM + o0 + l;  // col o0+l
#pragma unroll 4
  for (int kc = 0; kc < SEQ; kc += 4) {
    const int ka = kc + 2 * half;
    v2f a, bb;
    a.x  = abase[(size_t)(ka    ) * MAX_OUT_SEQ];
    a.y  = abase[(size_t)(ka + 1) * MAX_OUT_SEQ];
    bb.x = bbase[(size_t)(ka    ) * OUT_DIM];
    bb.y = bbase[(size_t)(ka + 1) * OUT_DIM];
    c = __builtin_amdgcn_wmma_f32_16x16x4_f32(false, a, false, bb,
                                              (short)0, c, false, false);
  }
  float* o = out + (size_t)(b * OUT_SEQ + t0 + half * 8) * OUT_DIM + o0 + l;
#pragma unroll
  for (int vg = 0; vg < 8; ++vg)
    o[(size_t)vg * OUT_DIM] = c[vg];

  // tuple's second element (out_seq) if the harness allocated room for it
  if (tile == 0 && lane == 0 && out_size > BATCH * OUT_SEQ * OUT_DIM)
    out[BATCH * OUT_SEQ * OUT_DIM] = (float)OUT_SEQ;
}

// ---------------------------------------------------------------------------
extern "C" void kernel_launch(void* const* d_in, const int* in_sizes, int n_in,
                              void* d_out, int out_size, void* d_ws, size_t ws_size,
                              hipStream_t stream) {
  const float* x     = (const float*)d_in[0];
  const float* M     = (const float*)d_in[1];
  const float* P     = (const float*)d_in[2];
  const float* Lnk   = (const float*)d_in[3];
  const float* gamma = (const float*)d_in[4];
  const float* beta  = (const float*)d_in[5];
  (void)in_sizes; (void)n_in; (void)ws_size;

  float* Z = (float*)d_ws;                          // 2048*256 f32 = 2 MB
  float* S = Z + (size_t)TOTAL_ROWS * OUT_DIM;      // 2 MB more

  gemm_xMt <<<256,        256, 0, stream>>>(x, M, Z);
  modrow   <<<TOTAL_ROWS, 256, 0, stream>>>(Z, P, gamma, beta, S);
  gemm_link<<<128,        256, 0, stream>>>(Lnk, S, (float*)d_out, out_size);
}